// EMAVQ_79353815761237
// MI455X (gfx1250) — compile-verified
//
#include <hip/hip_runtime.h>

// ---------------------------------------------------------------------------
// EMA-VQ eval forward for MI455X (gfx1250, wave32, WMMA).
// feats: (8,1024,256) f32  -> N=8192 rows, D=256
// emb:   (8192,256)  f32   -> V=8192 codes
// dist(n,v) = ||x||^2 - 2 x.e + ||e||^2 ; argmin over v ignores ||x||^2.
// GEMM 8192x8192x256 f32 via V_WMMA_F32_16X16X4_F32 (full f32 precision --
// bf16 mantissa is too coarse for the argmin gaps of this codebook).
// ---------------------------------------------------------------------------

typedef __attribute__((ext_vector_type(2))) float v2f;
typedef __attribute__((ext_vector_type(4))) float v4f;
typedef __attribute__((ext_vector_type(8))) float v8f;

#define N_ROWS   8192
#define N_CODES  8192
#define DIM      256
#define N_ELEMS  (N_ROWS * DIM)          // 2097152

// ---------------------------------------------------------------------------
// K1: ||e_v||^2 per code. One wave per code row: 32 lanes x 2 float4 loads.
// ---------------------------------------------------------------------------
__global__ __launch_bounds__(256) void vq_norm_kernel(
    const float* __restrict__ emb, float* __restrict__ normE)
{
    const int lane = threadIdx.x & 31;
    const int code = (blockIdx.x * blockDim.x + threadIdx.x) >> 5;
    const v4f* row = (const v4f*)(emb + (size_t)code * DIM);
    float s = 0.0f;
#pragma unroll
    for (int j = 0; j < 2; ++j) {
        v4f v = row[lane + 32 * j];
        s += v.x * v.x + v.y * v.y + v.z * v.z + v.w * v.w;
    }
#pragma unroll
    for (int off = 16; off >= 1; off >>= 1)
        s += __shfl_xor(s, off, 32);
    if (lane == 0) normE[code] = s;
}

// ---------------------------------------------------------------------------
// K2: main argmin kernel. One wave owns a 16-row tile of feats; A fragments
// (16x256) are held resident in 64 float2 regs. Loop over code tiles two at a
// time (two independent WMMA accumulation chains). f32 WMMA layout
// (16x16x4): lane L -> m = L&15, khalf = (L>>4)*2; A frag = {A[m][4j+khalf],
// A[m][4j+khalf+1]}, B frag = {E[n][4j+khalf], E[n][4j+khalf+1]} with n=L&15.
// C/D: vgpr g, lanes 0-15 -> (M=g, N=lane), lanes 16-31 -> (M=g+8, N=lane-16).
// ---------------------------------------------------------------------------
__global__ __launch_bounds__(256) void vq_argmin_kernel(
    const float* __restrict__ feats, const float* __restrict__ emb,
    const float* __restrict__ normE, int* __restrict__ idx_int,
    float* __restrict__ idx_f32)
{
    const int lane  = threadIdx.x & 31;
    const int wave  = threadIdx.x >> 5;
    const int m     = lane & 15;
    const int khalf = (lane >> 4) * 2;                 // 0 or 2
    const int row_base = (blockIdx.x * 8 + wave) * 16; // 512 row tiles total

    // Preload this wave's A tile: afrag[j] covers k = 4j .. 4j+3.
    v2f afrag[64];
    {
        const v2f* arow = (const v2f*)(feats + (size_t)(row_base + m) * DIM + khalf);
#pragma unroll
        for (int j = 0; j < 64; ++j)
            afrag[j] = arow[2 * j];
    }

    float best_s[8];
    int   best_i[8];
#pragma unroll
    for (int g = 0; g < 8; ++g) { best_s[g] = 3.4e38f; best_i[g] = 0; }

    for (int ct = 0; ct < N_CODES / 16; ct += 2) {
        const v2f* b0 = (const v2f*)(emb + (size_t)(ct * 16 + m) * DIM + khalf);
        const v2f* b1 = b0 + (16 * DIM) / 2;           // next 16 codes

        v8f acc0 = {};
        v8f acc1 = {};
#pragma unroll
        for (int j = 0; j < 64; ++j) {
            v2f bb0 = b0[2 * j];
            v2f bb1 = b1[2 * j];
            acc0 = __builtin_amdgcn_wmma_f32_16x16x4_f32(
                false, afrag[j], false, bb0, (short)0, acc0, false, false);
            acc1 = __builtin_amdgcn_wmma_f32_16x16x4_f32(
                false, afrag[j], false, bb1, (short)0, acc1, false, false);
        }

        const int   c0 = ct * 16 + m;
        const int   c1 = c0 + 16;
        const float n0 = normE[c0];
        const float n1 = normE[c1];
#pragma unroll
        for (int g = 0; g < 8; ++g) {
            float s0 = n0 - 2.0f * acc0[g];
            if (s0 < best_s[g]) { best_s[g] = s0; best_i[g] = c0; }
            float s1 = n1 - 2.0f * acc1[g];
            if (s1 < best_s[g]) { best_s[g] = s1; best_i[g] = c1; }
        }
    }

    // Min-reduce across the 16 lanes sharing a row (xor masks < 16 stay
    // inside each half-wave group). Tie-break: lowest code index.
#pragma unroll
    for (int g = 0; g < 8; ++g) {
        float s = best_s[g];
        int   i = best_i[g];
#pragma unroll
        for (int off = 8; off >= 1; off >>= 1) {
            float os = __shfl_xor(s, off, 32);
            int   oi = __shfl_xor(i, off, 32);
            if (os < s || (os == s && oi < i)) { s = os; i = oi; }
        }
        if ((lane & 15) == 0) {
            int row = row_base + g + ((lane >> 4) << 3);
            idx_int[row] = i;
            idx_f32[row] = (float)i;
        }
    }
}

// ---------------------------------------------------------------------------
// K3: quant = emb[idx] (== quant_st forward value), fused MSE accumulation.
// float4 per thread; wave-reduce then one atomicAdd per wave.
// ---------------------------------------------------------------------------
__global__ __launch_bounds__(256) void vq_gather_loss_kernel(
    const float* __restrict__ feats, const float* __restrict__ emb,
    const int* __restrict__ idx_int, float* __restrict__ quant_out,
    float* __restrict__ loss_acc)
{
    const int gid4 = blockIdx.x * blockDim.x + threadIdx.x;  // float4 index
    const int row  = gid4 >> 6;                              // 64 float4 per row
    const int dpos = gid4 & 63;

    const v4f q = ((const v4f*)(emb + (size_t)idx_int[row] * DIM))[dpos];
    const v4f f = ((const v4f*)feats)[gid4];
    ((v4f*)quant_out)[gid4] = q;

    v4f d;
    d.x = q.x - f.x; d.y = q.y - f.y; d.z = q.z - f.z; d.w = q.w - f.w;
    float s = d.x * d.x + d.y * d.y + d.z * d.z + d.w * d.w;
#pragma unroll
    for (int off = 16; off >= 1; off >>= 1)
        s += __shfl_xor(s, off, 32);
    if ((threadIdx.x & 31) == 0) atomicAdd(loss_acc, s);
}

__global__ void vq_zero_kernel(float* __restrict__ loss_acc) { *loss_acc = 0.0f; }

__global__ void vq_finalize_kernel(const float* __restrict__ loss_acc,
                                   float* __restrict__ loss_out)
{
    *loss_out = *loss_acc * (1.0f / (float)N_ELEMS);
}

// ---------------------------------------------------------------------------
// Launcher. d_out layout: [quant f32 x 2097152][idx-as-f32 x 8192][loss x 1].
// ws layout: [normE f32 x 8192][idx int x 8192][loss_acc f32 x 1].
// ---------------------------------------------------------------------------
extern "C" void kernel_launch(void* const* d_in, const int* in_sizes, int n_in,
                              void* d_out, int out_size, void* d_ws, size_t ws_size,
                              hipStream_t stream)
{
    (void)in_sizes; (void)n_in; (void)out_size; (void)ws_size;

    const float* feats = (const float*)d_in[0];
    const float* emb   = (const float*)d_in[1];

    float* out       = (float*)d_out;
    float* quant_out = out;                       // 2097152
    float* idx_f32   = out + N_ELEMS;             // 8192
    float* loss_out  = out + N_ELEMS + N_ROWS;    // 1

    float* normE     = (float*)d_ws;              // 8192 f32
    int*   idx_int   = (int*)(normE + N_CODES);   // 8192 i32
    float* loss_acc  = (float*)(idx_int + N_ROWS);

    // K1: code norms (8192 waves, 8 waves/block)
    vq_norm_kernel<<<N_CODES / 8, 256, 0, stream>>>(emb, normE);

    // K2: WMMA distance + argmin (512 row tiles, 8 waves/block)
    vq_argmin_kernel<<<(N_ROWS / 16) / 8, 256, 0, stream>>>(
        feats, emb, normE, idx_int, idx_f32);

    // K3/K4: gather + loss
    vq_zero_kernel<<<1, 1, 0, stream>>>(loss_acc);
    vq_gather_loss_kernel<<<(N_ELEMS / 4) / 256, 256, 0, stream>>>(
        feats, emb, idx_int, quant_out, loss_acc);
    vq_finalize_kernel<<<1, 1, 0, stream>>>(loss_acc, loss_out);
}